// TS_enc_34187939676176
// MI455X (gfx1250) — compile-verified
//
#include <hip/hip_runtime.h>
#include <hip/hip_bf16.h>

typedef float v2f __attribute__((ext_vector_type(2)));
typedef float v8f __attribute__((ext_vector_type(8)));

#define BATCH   262144
#define NE0     23      // edges in topology
#define NN0     24      // NE0 + 1 (appended "group" pseudo-edge)
#define NPOOL   14      // pooled groups (84/6)
#define NN1     14      // 13 new edges + group

// d_ws float layout
#define WS_WP   0                 // Wp  [84][72]  = POOLW @ (W1 * MASK1)
#define WS_BP   6048              // bp  [84]      = POOLW @ b1
#define WS_W2   6132              // W2m [168][84] = W2 * MASK2
#define WS_B2   (6132 + 14112)    // b2  [168]

// ---------------------------------------------------------------------------
// find_neighbor(edges, d=2) -> bitmask per edge-node (incl. appended group)
// ---------------------------------------------------------------------------
__device__ static void nb_masks(const int* ea, const int* eb, int n, unsigned* out) {
  int dist[NN0][NN0];
  for (int i = 0; i < n; ++i)
    for (int j = 0; j < n; ++j) {
      bool adj = (ea[i]==ea[j]) || (ea[i]==eb[j]) || (eb[i]==ea[j]) || (eb[i]==eb[j]);
      dist[i][j] = (i == j) ? 0 : (adj ? 1 : 100000);
    }
  for (int k = 0; k < n; ++k)
    for (int i = 0; i < n; ++i)
      for (int j = 0; j < n; ++j) {
        int v = dist[i][k] + dist[k][j];
        if (v < dist[i][j]) dist[i][j] = v;
      }
  for (int i = 0; i < n; ++i) {
    unsigned m = 0;
    for (int j = 0; j < n; ++j) if (dist[i][j] <= 2) m |= (1u << j);
    out[i] = m;
  }
  unsigned gp = out[0];
  for (int i = 0; i < n; ++i) if ((gp >> i) & 1u) out[i] |= (1u << n);
  out[n] = gp | (1u << n);
}

// ---------------------------------------------------------------------------
// Setup: build masks + pooling from TOPOLOGY, fuse into Wp/bp/W2m/b2 in ws
// ---------------------------------------------------------------------------
__global__ __launch_bounds__(256) void skel_setup(const float* __restrict__ W1, const float* __restrict__ b1,
                                                  const float* __restrict__ W2, const float* __restrict__ b2,
                                                  float* __restrict__ ws) {
  __shared__ unsigned nb0[NN0];
  __shared__ unsigned nb1[NN1];
  __shared__ int pp[NPOOL][2];
  __shared__ int plen[NPOOL];

  if (threadIdx.x == 0) {
    const int topo[24] = {0,0,0,0,1,2,3,4,5,6,7,8,9,9,9,12,13,14,16,17,18,19,20,21};
    int ea[NE0], eb[NE0];
    for (int i = 0; i < NE0; ++i) { ea[i] = topo[i+1]; eb[i] = i+1; }

    { unsigned tmp[NN0]; nb_masks(ea, eb, NE0, tmp);
      for (int i = 0; i < NN0; ++i) nb0[i] = tmp[i]; }

    int deg[32];
    for (int i = 0; i < 32; ++i) deg[i] = 0;
    for (int i = 0; i < NE0; ++i) { deg[ea[i]]++; deg[eb[i]]++; }

    // skeleton_pool DFS (pre-order, children in ascending edge index)
    int seqs[16][24]; int seqLen[16]; int nseq = 0;
    int stN[32]; int stSeq[32][24]; int stL[32]; int sp = 0;
    stN[0] = 0; stL[0] = 0; sp = 1;
    while (sp > 0) {
      --sp;
      int j = stN[sp]; int len = stL[sp];
      int seq[24];
      for (int t = 0; t < len; ++t) seq[t] = stSeq[sp][t];
      if (deg[j] > 2 && j != 0) {
        for (int t = 0; t < len; ++t) seqs[nseq][t] = seq[t];
        seqLen[nseq++] = len;
        len = 0;
      }
      if (deg[j] == 1) {
        for (int t = 0; t < len; ++t) seqs[nseq][t] = seq[t];
        seqLen[nseq++] = len;
        continue;
      }
      for (int idx = NE0 - 1; idx >= 0; --idx) {    // reversed push => pre-order pop
        if (ea[idx] == j) {
          stN[sp] = eb[idx];
          for (int t = 0; t < len; ++t) stSeq[sp][t] = seq[t];
          stSeq[sp][len] = idx; stL[sp] = len + 1; ++sp;
        }
      }
    }

    int np = 0, ne1 = 0;
    int e1a[NE0], e1b[NE0];
    for (int s = 0; s < nseq; ++s) {
      int L = seqLen[s]; const int* S = seqs[s]; int st = 0;
      if (L & 1) {
        pp[np][0] = S[0]; pp[np][1] = -1; plen[np] = 1;
        e1a[ne1] = ea[S[0]]; e1b[ne1] = eb[S[0]]; ++np; ++ne1; st = 1;
      }
      for (int i = st; i < L; i += 2) {
        pp[np][0] = S[i]; pp[np][1] = S[i+1]; plen[np] = 2;
        e1a[ne1] = ea[S[i]]; e1b[ne1] = eb[S[i+1]]; ++np; ++ne1;
      }
    }
    pp[np][0] = NE0; pp[np][1] = -1; plen[np] = 1; ++np;   // trailing [edge_num-1]

    { unsigned tmp[NN0]; nb_masks(e1a, e1b, ne1, tmp);
      for (int i = 0; i <= ne1; ++i) nb1[i] = tmp[i]; }
  }
  __syncthreads();

  float* Wp  = ws + WS_WP;
  float* bp  = ws + WS_BP;
  float* W2m = ws + WS_W2;
  float* b2o = ws + WS_B2;

  for (int idx = threadIdx.x; idx < 84 * 72; idx += 256) {
    int i = idx / 72, j = idx % 72;
    int p = i / 6, r = i % 6, jn = j / 3;
    int L = plen[p];
    float s = 0.f;
    for (int t = 0; t < L; ++t) {
      int e = pp[p][t];
      if ((nb0[e] >> jn) & 1u) s += W1[(e * 6 + r) * 72 + j];
    }
    Wp[idx] = s / (float)L;
  }
  for (int idx = threadIdx.x; idx < 84; idx += 256) {
    int p = idx / 6, r = idx % 6;
    int L = plen[p];
    float s = 0.f;
    for (int t = 0; t < L; ++t) s += b1[pp[p][t] * 6 + r];
    bp[idx] = s / (float)L;
  }
  for (int idx = threadIdx.x; idx < 168 * 84; idx += 256) {
    int i = idx / 84, j = idx % 84;
    int g = i / 12, jn = j / 6;
    W2m[idx] = ((nb1[g] >> jn) & 1u) ? W2[idx] : 0.f;
  }
  for (int idx = threadIdx.x; idx < 168; idx += 256) b2o[idx] = b2[idx];
}

// ---------------------------------------------------------------------------
// GEMM1: o1 = leaky(x[B,72] @ Wp^T[72,84] + bp)  via V_WMMA_F32_16X16X4_F32
// ---------------------------------------------------------------------------
#define S1 74   // LDS row stride (floats): b64-aligned, conflict-free for 16 lanes
__global__ __launch_bounds__(128) void skel_gemm1(const float* __restrict__ x,
                                                  const float* __restrict__ ws,
                                                  float* __restrict__ o1) {
  __shared__ float sW[96 * S1];
  __shared__ float sb[96];
  const float* Wp = ws + WS_WP;
  const float* bp = ws + WS_BP;

  for (int i = threadIdx.x; i < 96 * S1; i += 128) sW[i] = 0.f;
  for (int i = threadIdx.x; i < 96; i += 128) sb[i] = (i < 84) ? bp[i] : 0.f;
  __syncthreads();
  for (int i = threadIdx.x; i < 84 * 72; i += 128) {
    int n = i / 72, k = i % 72;
    sW[n * S1 + k] = Wp[i];
  }
  __syncthreads();

  const int lane  = threadIdx.x & 31;
  const int wave  = threadIdx.x >> 5;
  const int row0  = blockIdx.x * 64 + wave * 16;
  const int nlane = lane & 15;
  const int khalf = (lane >> 4) * 2;
  const int mlane = row0 + nlane;

  v2f A[18];
#pragma unroll
  for (int t = 0; t < 18; ++t)
    A[t] = *(const v2f*)(x + (size_t)mlane * 72 + t * 4 + khalf);

  v8f acc[6] = {};
#pragma unroll
  for (int t = 0; t < 18; ++t) {
#pragma unroll
    for (int n = 0; n < 6; ++n) {
      v2f b = *(const v2f*)(&sW[(n * 16 + nlane) * S1 + t * 4 + khalf]);
      acc[n] = __builtin_amdgcn_wmma_f32_16x16x4_f32(false, A[t], false, b,
                                                     (short)0, acc[n], false, false);
    }
  }

#pragma unroll
  for (int n = 0; n < 6; ++n) {
    int col = n * 16 + nlane;
    if (col < 84) {
      float bias = sb[col];
#pragma unroll
      for (int r = 0; r < 8; ++r) {
        int m = row0 + r + (lane >> 4) * 8;
        float v = acc[n][r] + bias;
        v = (v >= 0.f) ? v : 0.2f * v;
        o1[(size_t)m * 84 + col] = v;
      }
    }
  }
}

// ---------------------------------------------------------------------------
// GEMM2: o2 = leaky(o1[B,84] @ W2m^T[84,168] + b2)
// ---------------------------------------------------------------------------
#define S2 86
__global__ __launch_bounds__(128) void skel_gemm2(const float* __restrict__ o1,
                                                  const float* __restrict__ ws,
                                                  float* __restrict__ o2) {
  __shared__ float sW[176 * S2];
  __shared__ float sb[176];
  const float* W2m = ws + WS_W2;
  const float* b2  = ws + WS_B2;

  for (int i = threadIdx.x; i < 176 * S2; i += 128) sW[i] = 0.f;
  for (int i = threadIdx.x; i < 176; i += 128) sb[i] = (i < 168) ? b2[i] : 0.f;
  __syncthreads();
  for (int i = threadIdx.x; i < 168 * 84; i += 128) {
    int n = i / 84, k = i % 84;
    sW[n * S2 + k] = W2m[i];
  }
  __syncthreads();

  const int lane  = threadIdx.x & 31;
  const int wave  = threadIdx.x >> 5;
  const int row0  = blockIdx.x * 64 + wave * 16;
  const int nlane = lane & 15;
  const int khalf = (lane >> 4) * 2;
  const int mlane = row0 + nlane;

  v2f A[21];
#pragma unroll
  for (int t = 0; t < 21; ++t)
    A[t] = *(const v2f*)(o1 + (size_t)mlane * 84 + t * 4 + khalf);

  v8f acc[11] = {};
#pragma unroll
  for (int t = 0; t < 21; ++t) {
#pragma unroll
    for (int n = 0; n < 11; ++n) {
      v2f b = *(const v2f*)(&sW[(n * 16 + nlane) * S2 + t * 4 + khalf]);
      acc[n] = __builtin_amdgcn_wmma_f32_16x16x4_f32(false, A[t], false, b,
                                                     (short)0, acc[n], false, false);
    }
  }

#pragma unroll
  for (int n = 0; n < 11; ++n) {
    int col = n * 16 + nlane;
    if (col < 168) {
      float bias = sb[col];
#pragma unroll
      for (int r = 0; r < 8; ++r) {
        int m = row0 + r + (lane >> 4) * 8;
        float v = acc[n][r] + bias;
        v = (v >= 0.f) ? v : 0.2f * v;
        o2[(size_t)m * 168 + col] = v;
      }
    }
  }
}

// ---------------------------------------------------------------------------
extern "C" void kernel_launch(void* const* d_in, const int* in_sizes, int n_in,
                              void* d_out, int out_size, void* d_ws, size_t ws_size,
                              hipStream_t stream) {
  (void)in_sizes; (void)n_in; (void)out_size; (void)ws_size;
  const float* x  = (const float*)d_in[0];
  const float* W1 = (const float*)d_in[1];
  const float* b1 = (const float*)d_in[2];
  const float* W2 = (const float*)d_in[3];
  const float* b2 = (const float*)d_in[4];

  float* out = (float*)d_out;
  float* ws  = (float*)d_ws;
  float* o1  = out + (size_t)BATCH * 72;
  float* o2  = o1 + (size_t)BATCH * 84;

  // output[0]: passthrough copy of x (graph-capturable d2d async copy)
  hipMemcpyAsync(out, x, (size_t)BATCH * 72 * sizeof(float),
                 hipMemcpyDeviceToDevice, stream);

  skel_setup<<<1, 256, 0, stream>>>(W1, b1, W2, b2, ws);
  skel_gemm1<<<BATCH / 64, 128, 0, stream>>>(x, ws, o1);
  skel_gemm2<<<BATCH / 64, 128, 0, stream>>>(o1, ws, o2);
}